// PointPropagation_73323681677767
// MI455X (gfx1250) — compile-verified
//
#include <hip/hip_runtime.h>
#include <hip/hip_bf16.h>

#define N_  4
#define C_  64
#define H_  400
#define W_  400
#define HW_ (H_ * W_)
#define NP_ (N_ * HW_)

typedef __attribute__((ext_vector_type(2))) float v2f;
typedef __attribute__((ext_vector_type(8))) float v8f;

// ---------------------------------------------------------------------------
// Kernel 1: per-pixel channel projection via V_WMMA_F32_16X16X4_F32 (K=64 as
// 16 accumulating K=4 steps). Each wave produces a 16-pixel x 16-output tile;
// outputs 0..4 = center0, center1, step0, step1, prob-logit. Then per-pixel
// lanes compute offset -> tgt index (round-half-even, clamp max, wrap mod)
// and sigmoid(prob).
// ---------------------------------------------------------------------------
__global__ __launch_bounds__(256) void pp_proj_kernel(
    const float* __restrict__ feat, const float* __restrict__ wc,
    const float* __restrict__ ws, const float* __restrict__ wp,
    float* __restrict__ probbuf, int* __restrict__ tgtbuf) {
  __shared__ float Wlds[C_ * 16];     // [c][o] weight matrix, cols >=5 are zero
  __shared__ float Tile[8 * 16 * 8];  // [wave][pixel][out(5, padded to 8)]

  // Build the 64x16 weight matrix in LDS (only 5 columns non-zero).
  for (int idx = threadIdx.x; idx < C_ * 16; idx += 256) {
    int c = idx >> 4, o = idx & 15;
    float v = 0.0f;
    if (o == 0)      v = wc[c];
    else if (o == 1) v = wc[C_ + c];
    else if (o == 2) v = ws[c];
    else if (o == 3) v = ws[C_ + c];
    else if (o == 4) v = wp[c];
    Wlds[idx] = v;
  }
  __syncthreads();

  const int wave = threadIdx.x >> 5;
  const int lane = threadIdx.x & 31;
  const int half = lane >> 4;   // which 16-lane half
  const int lm   = lane & 15;

  const int n       = blockIdx.x / (HW_ / 128);
  const int pixbase = (blockIdx.x % (HW_ / 128)) * 128 + wave * 16;
  const float* fb   = feat + (size_t)n * C_ * HW_;

  // A (16x4 f32): lane half selects K pair {0,1} vs {2,3}; row M = lm = pixel.
  // B (4x16 f32): same K split across halves; column N = lm = output.
  v8f acc = {};
#pragma unroll
  for (int k = 0; k < 16; ++k) {
    const int c0 = 4 * k + 2 * half;
    const int p  = pixbase + lm;
    v2f a, b;
    a.x = fb[(size_t)c0 * HW_ + p];
    a.y = fb[(size_t)(c0 + 1) * HW_ + p];
    b.x = Wlds[(4 * k + 2 * half) * 16 + lm];
    b.y = Wlds[(4 * k + 2 * half + 1) * 16 + lm];
    acc = __builtin_amdgcn_wmma_f32_16x16x4_f32(false, a, false, b, (short)0,
                                                acc, false, false);
  }

  // D layout: VGPR v holds (M = v + 8*half, N = lm). Keep the 5 used columns.
  if (lm < 5) {
#pragma unroll
    for (int v = 0; v < 8; ++v) {
      const int m = v + 8 * half;
      Tile[wave * 128 + m * 8 + lm] = acc[v];
    }
  }
  __syncthreads();

  if (lane < 16) {
    const int p  = pixbase + lane;
    const float c0 = Tile[wave * 128 + lane * 8 + 0];
    const float c1 = Tile[wave * 128 + lane * 8 + 1];
    const float s0 = Tile[wave * 128 + lane * 8 + 2];
    const float s1 = Tile[wave * 128 + lane * 8 + 3];
    const float pr = Tile[wave * 128 + lane * 8 + 4];

    const float off0 = c0 * fmaxf(s0, 0.0f);  // STRIDE == 1.0
    const float off1 = c1 * fmaxf(s1, 0.0f);
    const float gh = (float)(p / W_);
    const float gw = (float)(p % W_);
    // jnp.round == round half to even == rintf (default rounding mode)
    const float th = fminf(rintf(gh + off0), (float)(H_ - 1));
    const float tw = fminf(rintf(gw + off1), (float)(W_ - 1));
    int t = (int)(th * (float)W_ + tw);       // exact: |values| < 2^24
    int m = t % HW_;
    if (m < 0) m += HW_;                      // python-style mod (wrap neg)

    probbuf[n * HW_ + p] = 1.0f / (1.0f + __expf(-pr));
    tgtbuf[n * HW_ + p]  = m;
  }
}

// ---------------------------------------------------------------------------
// Kernel 2: init scatter buffer to -1 (identity sentinel).
// ---------------------------------------------------------------------------
__global__ __launch_bounds__(256) void pp_init_kernel(int* __restrict__ samp) {
  int i = blockIdx.x * 256 + threadIdx.x;
  if (i < NP_) samp[i] = -1;
}

// ---------------------------------------------------------------------------
// Kernel 3: scatter source index: samp[n][tgt[n][j]] = j. atomicMax makes the
// duplicate-index winner deterministic (largest j).
// ---------------------------------------------------------------------------
__global__ __launch_bounds__(256) void pp_scatter_kernel(
    const int* __restrict__ tgtbuf, int* __restrict__ samp) {
  int gi = blockIdx.x * 256 + threadIdx.x;
  if (gi >= NP_) return;
  const int n = gi / HW_, j = gi % HW_;
  const int m = tgtbuf[gi];
  atomicMax(&samp[n * HW_ + m], j);
}

// ---------------------------------------------------------------------------
// Kernel 4: fused gather + double sigmoid blend. Since H==W and the sampled
// coordinates are exact integers, the bilinear grid-sample reduces exactly to
// a gather at the transposed flat index q = (s%W)*W + s/W.
// One thread per (n, pixel); loop over the 64 channels (coalesced streams,
// gather stays L2-resident: one (n,c) plane = 640 KB << 192 MB L2).
// ---------------------------------------------------------------------------
__global__ __launch_bounds__(256) void pp_fuse_kernel(
    const float* __restrict__ feat, const float* __restrict__ probbuf,
    const int* __restrict__ samp, float* __restrict__ out) {
  int gi = blockIdx.x * 256 + threadIdx.x;
  if (gi >= NP_) return;
  const int n = gi / HW_, i = gi % HW_;
  int s = samp[gi];
  if (s < 0) s = i;  // identity where nothing scattered
  const int q = (s % W_) * W_ + (s / W_);
  const float p = probbuf[gi];
  const float* fb = feat + (size_t)n * C_ * HW_;
  float* ob       = out  + (size_t)n * C_ * HW_;
#pragma unroll 4
  for (int c = 0; c < C_; ++c) {
    const float f = fb[(size_t)c * HW_ + i];
    const float g = fb[(size_t)c * HW_ + q];
    const float proped = g * p + f * (1.0f - p);
    ob[(size_t)c * HW_ + i] = f * p + proped * (1.0f - p);
  }
}

extern "C" void kernel_launch(void* const* d_in, const int* in_sizes, int n_in,
                              void* d_out, int out_size, void* d_ws,
                              size_t ws_size, hipStream_t stream) {
  (void)in_sizes; (void)n_in; (void)out_size; (void)ws_size;
  const float* feat = (const float*)d_in[0];  // [4,64,400,400]
  const float* wc   = (const float*)d_in[1];  // [2,64]
  const float* ws   = (const float*)d_in[2];  // [2,64]
  const float* wp   = (const float*)d_in[3];  // [1,64]
  float* out = (float*)d_out;

  float* probbuf = (float*)d_ws;                                  // NP_ f32
  int*   tgtbuf  = (int*)((char*)d_ws + (size_t)NP_ * 4);         // NP_ i32
  int*   sampbuf = (int*)((char*)d_ws + (size_t)NP_ * 8);         // NP_ i32

  // 128 pixels per block (8 waves x 16-pixel WMMA tiles); HW_ % 128 == 0.
  pp_proj_kernel<<<NP_ / 128, 256, 0, stream>>>(feat, wc, ws, wp, probbuf,
                                                tgtbuf);
  pp_init_kernel<<<(NP_ + 255) / 256, 256, 0, stream>>>(sampbuf);
  pp_scatter_kernel<<<(NP_ + 255) / 256, 256, 0, stream>>>(tgtbuf, sampbuf);
  pp_fuse_kernel<<<(NP_ + 255) / 256, 256, 0, stream>>>(feat, probbuf, sampbuf,
                                                        out);
}